// OHEML1Loss_32220844654661
// MI455X (gfx1250) — compile-verified
//
#include <hip/hip_runtime.h>
#include <hip/hip_bf16.h>

// OHEM smooth-L1 loss for (64, 256,256,4) f32 inputs.
// Exact per-row top-k threshold via 3-level radix select on float bits
// (11/11/9), then a masked sum. Ties at the threshold are all negatives
// (y_true==0) so each contributes the identical smooth_l1(T) -> add
// (num_neg - cnt_gt) * f(T) analytically. All 4 data sweeps are L2-resident
// on MI455X (134 MB working set < 192 MB L2). Cross-lane reductions use
// V_WMMA_F32_16X16X4_F32 (wave32, A=partials 16x4, B=ones) + ds_swizzle.
// LDS histograms are 4-way privatized (32 KB of the WGP's 320 KB) to cut
// inter-wave same-address DS-atomic serialization on hot bins.

#define BATCH 64
#define NPR   262144            // elements per row = 256*256*4
#define NPR4  (NPR / 4)
#define RBLK  16                // streaming blocks per row
#define CHUNK4 (NPR4 / RBLK)    // 4096 float4 per block
#define BINS0 2048
#define BINS2 512
#define NCOPY 4                 // LDS histogram privatization copies
#define SENT  0xFFFFFFFFu

// ds_swizzle group-of-32 XOR patterns: offset = (xor_mask<<10) | and_mask(0x1F)
#define SWZ_X16 0x401F
#define SWZ_X8  0x201F
#define SWZ_X4  0x101F
#define SWZ_X2  0x081F
#define SWZ_X1  0x041F

typedef __attribute__((ext_vector_type(2))) float v2f;
typedef __attribute__((ext_vector_type(8))) float v8f;

__device__ __forceinline__ float smoothl1(float a) {       // a = |d|
  return (a < 1.0f) ? 0.5f * a * a : (a - 0.5f);
}

__device__ __forceinline__ unsigned keyOf(float p, float t) {
  float a = fabsf(p - t);
  return (t != 0.0f) ? 0u : __float_as_uint(a);            // loss_c bits
}

// Wave32 sum of one float per lane via V_WMMA_F32_16X16X4_F32.
// A[m][0]=p[m], A[m][2]=p[m+16], A[m][1]=A[m][3]=0; B = ones.
// D[m][n] = p[m]+p[m+16]; sum of 8 D VGPRs gives half-totals; xor-16 combines.
__device__ __forceinline__ float wave_sum_f32(float v) {
  v2f a; a.x = v;    a.y = 0.0f;
  v2f b; b.x = 1.0f; b.y = 1.0f;
  v8f c = {};
  v8f d = __builtin_amdgcn_wmma_f32_16x16x4_f32(false, a, false, b,
                                                (short)0, c, false, false);
  float t = d[0] + d[1] + d[2] + d[3] + d[4] + d[5] + d[6] + d[7];
  t += __int_as_float(__builtin_amdgcn_ds_swizzle(__float_as_int(t), SWZ_X16));
  return t;                                                // total in all lanes
}

__device__ __forceinline__ unsigned wave_sum_u32(unsigned v) {
  v += (unsigned)__builtin_amdgcn_ds_swizzle((int)v, SWZ_X16);
  v += (unsigned)__builtin_amdgcn_ds_swizzle((int)v, SWZ_X8);
  v += (unsigned)__builtin_amdgcn_ds_swizzle((int)v, SWZ_X4);
  v += (unsigned)__builtin_amdgcn_ds_swizzle((int)v, SWZ_X2);
  v += (unsigned)__builtin_amdgcn_ds_swizzle((int)v, SWZ_X1);
  return v;
}

// Pass 1: level-0 histogram (bits 30:20), per-row pos count, pos smooth-L1 sum.
__global__ __launch_bounds__(256) void k_pass1(
    const float* __restrict__ yp, const float* __restrict__ yt,
    unsigned* __restrict__ hist, unsigned* __restrict__ num_pos,
    float* __restrict__ posPart) {
  __shared__ unsigned lh[NCOPY][BINS0];                    // 32 KB
  __shared__ float    wsum[8];
  __shared__ unsigned wcnt[8];
  const int row = blockIdx.y, tid = threadIdx.x;
  const int cp = (tid >> 5) & (NCOPY - 1);                 // copy per wave
  for (int i = tid; i < NCOPY * BINS0; i += 256) (&lh[0][0])[i] = 0u;
  __syncthreads();
  const float4* p4 = (const float4*)yp;
  const float4* t4 = (const float4*)yt;
  const long base = (long)row * NPR4 + (long)blockIdx.x * CHUNK4;
  float psum = 0.0f; unsigned pcnt = 0u;
#define P1(PC, TC) { float a_ = fabsf((PC) - (TC));                        \
    if ((TC) != 0.0f) { pcnt++; psum += smoothl1(a_);                      \
                        atomicAdd(&lh[cp][0], 1u); }                       \
    else { atomicAdd(&lh[cp][(__float_as_uint(a_) >> 20) & 0x7FFu], 1u); } }
  for (int i = tid; i < CHUNK4; i += 256) {
    float4 p = p4[base + i]; float4 t = t4[base + i];
    P1(p.x, t.x) P1(p.y, t.y) P1(p.z, t.z) P1(p.w, t.w)
  }
#undef P1
  __syncthreads();
  for (int i = tid; i < BINS0; i += 256) {
    unsigned c = lh[0][i] + lh[1][i] + lh[2][i] + lh[3][i];
    if (c) atomicAdd(&hist[row * BINS0 + i], c);
  }
  float    ws = wave_sum_f32(psum);
  unsigned wc = wave_sum_u32(pcnt);
  const int lane = tid & 31, wv = tid >> 5;
  if (lane == 0) { wsum[wv] = ws; wcnt[wv] = wc; }
  __syncthreads();
  if (tid == 0) {
    float s = 0.0f; unsigned c = 0u;
    for (int w = 0; w < 8; ++w) { s += wsum[w]; c += wcnt[w]; }
    posPart[row * RBLK + blockIdx.x] = s;
    if (c) atomicAdd(&num_pos[row], c);
  }
}

// Pass 2: level-1 histogram (bits 19:9) restricted to level-0 bin beta0[row].
__global__ __launch_bounds__(256) void k_pass2(
    const float* __restrict__ yp, const float* __restrict__ yt,
    const unsigned* __restrict__ beta0, unsigned* __restrict__ hist) {
  __shared__ unsigned lh[NCOPY][BINS0];
  const int row = blockIdx.y, tid = threadIdx.x;
  const int cp = (tid >> 5) & (NCOPY - 1);
  const unsigned b0 = beta0[row];
  for (int i = tid; i < NCOPY * BINS0; i += 256) (&lh[0][0])[i] = 0u;
  __syncthreads();
  const float4* p4 = (const float4*)yp;
  const float4* t4 = (const float4*)yt;
  const long base = (long)row * NPR4 + (long)blockIdx.x * CHUNK4;
#define P2(PC, TC) { unsigned k_ = keyOf((PC), (TC));                      \
    if (((k_ >> 20) & 0x7FFu) == b0)                                       \
      atomicAdd(&lh[cp][(k_ >> 9) & 0x7FFu], 1u); }
  for (int i = tid; i < CHUNK4; i += 256) {
    float4 p = p4[base + i]; float4 t = t4[base + i];
    P2(p.x, t.x) P2(p.y, t.y) P2(p.z, t.z) P2(p.w, t.w)
  }
#undef P2
  __syncthreads();
  for (int i = tid; i < BINS0; i += 256) {
    unsigned c = lh[0][i] + lh[1][i] + lh[2][i] + lh[3][i];
    if (c) atomicAdd(&hist[row * BINS0 + i], c);
  }
}

// Pass 3: level-2 histogram (bits 8:0) restricted to (beta0,beta1).
__global__ __launch_bounds__(256) void k_pass3(
    const float* __restrict__ yp, const float* __restrict__ yt,
    const unsigned* __restrict__ beta0, const unsigned* __restrict__ beta1,
    unsigned* __restrict__ hist) {
  __shared__ unsigned lh[NCOPY][BINS2];                    // 8 KB
  const int row = blockIdx.y, tid = threadIdx.x;
  const int cp = (tid >> 5) & (NCOPY - 1);
  const unsigned hi = (beta0[row] << 11) | beta1[row];     // bits 31:9 target
  for (int i = tid; i < NCOPY * BINS2; i += 256) (&lh[0][0])[i] = 0u;
  __syncthreads();
  const float4* p4 = (const float4*)yp;
  const float4* t4 = (const float4*)yt;
  const long base = (long)row * NPR4 + (long)blockIdx.x * CHUNK4;
#define P3(PC, TC) { unsigned k_ = keyOf((PC), (TC));                      \
    if ((k_ >> 9) == hi) atomicAdd(&lh[cp][k_ & 0x1FFu], 1u); }
  for (int i = tid; i < CHUNK4; i += 256) {
    float4 p = p4[base + i]; float4 t = t4[base + i];
    P3(p.x, t.x) P3(p.y, t.y) P3(p.z, t.z) P3(p.w, t.w)
  }
#undef P3
  __syncthreads();
  for (int i = tid; i < BINS2; i += 256) {
    unsigned c = lh[0][i] + lh[1][i] + lh[2][i] + lh[3][i];
    if (c) atomicAdd(&hist[row * BINS2 + i], c);
  }
}

// Pass 4: sum smooth_l1 over elements with key > T[row] (strict negatives).
__global__ __launch_bounds__(256) void k_pass4(
    const float* __restrict__ yp, const float* __restrict__ yt,
    const unsigned* __restrict__ Tbits, float* __restrict__ negPart) {
  __shared__ float wsum[8];
  const int row = blockIdx.y, tid = threadIdx.x;
  const unsigned T = Tbits[row];
  const float4* p4 = (const float4*)yp;
  const float4* t4 = (const float4*)yt;
  const long base = (long)row * NPR4 + (long)blockIdx.x * CHUNK4;
  float nsum = 0.0f;
#define P4(PC, TC) { unsigned k_ = keyOf((PC), (TC));                      \
    if (k_ > T) nsum += smoothl1(__uint_as_float(k_)); }
  for (int i = tid; i < CHUNK4; i += 256) {
    float4 p = p4[base + i]; float4 t = t4[base + i];
    P4(p.x, t.x) P4(p.y, t.y) P4(p.z, t.z) P4(p.w, t.w)
  }
#undef P4
  float ws = wave_sum_f32(nsum);
  const int lane = tid & 31, wv = tid >> 5;
  if (lane == 0) wsum[wv] = ws;
  __syncthreads();
  if (tid == 0) {
    float s = 0.0f;
    for (int w = 0; w < 8; ++w) s += wsum[w];
    negPart[row * RBLK + blockIdx.x] = s;
  }
}

// Radix-select scans: one thread per row, descending cumulative count.
__global__ void k_scan0(const unsigned* __restrict__ hist,
                        const unsigned* __restrict__ num_pos,
                        unsigned* __restrict__ num_neg,
                        unsigned* __restrict__ beta0, unsigned* __restrict__ rem0) {
  const int r = threadIdx.x;
  if (r >= BATCH) return;
  unsigned np = num_pos[r];
  unsigned nn = 2u * np; if (nn > (unsigned)(NPR - 1)) nn = (unsigned)(NPR - 1);
  num_neg[r] = nn;
  unsigned beta = SENT, rem = 0u;
  if (nn) {
    unsigned cum = 0u;
    for (int b = BINS0 - 1; b >= 0; --b) {
      unsigned c = hist[r * BINS0 + b];
      if (cum + c >= nn) { beta = (unsigned)b; rem = nn - cum; break; }
      cum += c;
    }
  }
  beta0[r] = beta; rem0[r] = rem;
}

__global__ void k_scan1(const unsigned* __restrict__ hist,
                        const unsigned* __restrict__ remIn,
                        const unsigned* __restrict__ betaPrev,
                        unsigned* __restrict__ betaOut, unsigned* __restrict__ remOut) {
  const int r = threadIdx.x;
  if (r >= BATCH) return;
  unsigned beta = SENT, rem = 0u;
  if (betaPrev[r] != SENT) {
    const unsigned target = remIn[r];     // >= 1
    unsigned cum = 0u;
    for (int b = BINS0 - 1; b >= 0; --b) {
      unsigned c = hist[r * BINS0 + b];
      if (cum + c >= target) { beta = (unsigned)b; rem = target - cum; break; }
      cum += c;
    }
  }
  betaOut[r] = beta; remOut[r] = rem;
}

__global__ void k_scan2(const unsigned* __restrict__ hist,
                        const unsigned* __restrict__ remIn,
                        const unsigned* __restrict__ beta0,
                        const unsigned* __restrict__ beta1,
                        unsigned* __restrict__ Tbits, unsigned* __restrict__ k2v) {
  const int r = threadIdx.x;
  if (r >= BATCH) return;
  unsigned T = SENT, k2 = 0u;
  if (beta1[r] != SENT) {
    const unsigned target = remIn[r];
    unsigned cum = 0u;
    for (int b = BINS2 - 1; b >= 0; --b) {
      unsigned c = hist[r * BINS2 + b];
      if (cum + c >= target) {
        T  = (beta0[r] << 20) | (beta1[r] << 9) | (unsigned)b;
        k2 = target - cum;                // elements needed at exact value T
        break;
      }
      cum += c;
    }
  }
  Tbits[r] = T; k2v[r] = k2;
}

// Final combine: deterministic fixed-order reduction + tie terms + scalar loss.
__global__ __launch_bounds__(256) void k_final(
    const float* __restrict__ posPart, const float* __restrict__ negPart,
    const unsigned* __restrict__ num_pos, const unsigned* __restrict__ num_neg,
    const unsigned* __restrict__ Tbits, const unsigned* __restrict__ k2v,
    float* __restrict__ out) {
  __shared__ float red[256];
  __shared__ float posSumS;
  const int tid = threadIdx.x;
  const int NP = BATCH * RBLK;

  float s = 0.0f;
  for (int i = tid; i < NP; i += 256) s += posPart[i];
  red[tid] = s; __syncthreads();
  for (int o = 128; o > 0; o >>= 1) { if (tid < o) red[tid] += red[tid + o]; __syncthreads(); }
  if (tid == 0) posSumS = red[0];
  __syncthreads();

  s = 0.0f;
  for (int i = tid; i < NP; i += 256) s += negPart[i];
  red[tid] = s; __syncthreads();
  for (int o = 128; o > 0; o >>= 1) { if (tid < o) red[tid] += red[tid + o]; __syncthreads(); }

  if (tid == 0) {
    float negSum = red[0];
    float posSum = posSumS;
    unsigned pc = 0u, nc = 0u;
    for (int r = 0; r < BATCH; ++r) {
      pc += num_pos[r];
      nc += num_neg[r];
      unsigned kk = k2v[r];
      if (kk) negSum += (float)kk * smoothl1(__uint_as_float(Tbits[r]));
    }
    out[0] = 2.0f * (posSum / (float)pc) + (negSum / (float)nc);
  }
}

extern "C" void kernel_launch(void* const* d_in, const int* in_sizes, int n_in,
                              void* d_out, int out_size, void* d_ws, size_t ws_size,
                              hipStream_t stream) {
  (void)in_sizes; (void)n_in; (void)out_size; (void)ws_size;
  const float* yp = (const float*)d_in[0];
  const float* yt = (const float*)d_in[1];

  char* ws = (char*)d_ws;
  unsigned* hist    = (unsigned*)ws;                       // 64*2048 u32 = 512 KB (reused)
  unsigned* num_pos = (unsigned*)(ws + (size_t)BATCH * BINS0 * sizeof(unsigned));
  unsigned* num_neg = num_pos + 64;
  unsigned* beta0   = num_neg + 64;
  unsigned* rem0    = beta0 + 64;
  unsigned* beta1   = rem0 + 64;
  unsigned* rem1    = beta1 + 64;
  unsigned* Tbits   = rem1 + 64;
  unsigned* k2v     = Tbits + 64;
  float*    posPart = (float*)(k2v + 64);                  // 1024 f32
  float*    negPart = posPart + BATCH * RBLK;              // 1024 f32

  const dim3 grid(RBLK, BATCH), blk(256);

  hipMemsetAsync(hist, 0, (size_t)BATCH * BINS0 * sizeof(unsigned), stream);
  hipMemsetAsync(num_pos, 0, 64 * sizeof(unsigned), stream);
  k_pass1<<<grid, blk, 0, stream>>>(yp, yt, hist, num_pos, posPart);
  k_scan0<<<1, 64, 0, stream>>>(hist, num_pos, num_neg, beta0, rem0);

  hipMemsetAsync(hist, 0, (size_t)BATCH * BINS0 * sizeof(unsigned), stream);
  k_pass2<<<grid, blk, 0, stream>>>(yp, yt, beta0, hist);
  k_scan1<<<1, 64, 0, stream>>>(hist, rem0, beta0, beta1, rem1);

  hipMemsetAsync(hist, 0, (size_t)BATCH * BINS2 * sizeof(unsigned), stream);
  k_pass3<<<grid, blk, 0, stream>>>(yp, yt, beta0, beta1, hist);
  k_scan2<<<1, 64, 0, stream>>>(hist, rem1, beta0, beta1, Tbits, k2v);

  k_pass4<<<grid, blk, 0, stream>>>(yp, yt, Tbits, negPart);
  k_final<<<1, 256, 0, stream>>>(posPart, negPart, num_pos, num_neg, Tbits, k2v,
                                 (float*)d_out);
}